// SoftCLDiceLoss_59992103190581
// MI455X (gfx1250) — compile-verified
//
#include <hip/hip_runtime.h>
#include <hip/hip_bf16.h>

typedef __attribute__((ext_vector_type(2))) float v2f;
typedef __attribute__((ext_vector_type(8))) float v8f;

#define W_   192
#define H_   192
#define D_   128
#define BC_  4
#define SH_  192
#define SD_  (192*192)
#define VOL  ((size_t)D_ * SD_)     /* 4,718,592 per (b,c) */
#define VOLS (VOL * BC_)            /* 18,874,368 total    */

#define TW 64
#define TH 8
#define TD 8
#define LW (TW+2)   /* 66 */
#define LH (TH+2)   /* 10 */
#define LD (TD+2)   /* 10 */
#define LDS_N (LD*LH*LW) /* 6600 floats = 26.4 KB */

#define NBLK 128
#define CHUNK (VOL / NBLK)  /* 36864 */

__device__ __forceinline__ float fminmax(float a, float b, bool ismin) {
    return ismin ? fminf(a, b) : fmaxf(a, b);
}

// CDNA5 async copy: global -> LDS, no VGPR staging, tracked by ASYNCcnt.
// VDST operand = per-lane wave-relative LDS byte offset (low 32 bits of the
// generic shared pointer); VADDR = 64-bit global address; "off" = no SADDR.
__device__ __forceinline__ void async_g2lds_b32(const float* gp, float* lp) {
    unsigned off = (unsigned)(unsigned long long)lp;
    asm volatile("global_load_async_to_lds_b32 %0, %1, off"
                 :: "v"(off), "v"(gp)
                 : "memory");
}
__device__ __forceinline__ void wait_asynccnt0() {
    asm volatile("s_wait_asynccnt 0" ::: "memory");
}

// ---------------------------------------------------------------------------
// Elementwise sigmoid, float4-vectorized
// ---------------------------------------------------------------------------
__global__ __launch_bounds__(256) void k_sigmoid(const float4* __restrict__ x,
                                                 float4* __restrict__ y) {
    size_t i = (size_t)blockIdx.x * 256 + threadIdx.x;
    float4 v = x[i];
    float4 r;
    r.x = 1.0f / (1.0f + expf(-v.x));
    r.y = 1.0f / (1.0f + expf(-v.y));
    r.z = 1.0f / (1.0f + expf(-v.z));
    r.w = 1.0f / (1.0f + expf(-v.w));
    y[i] = r;
}

// ---------------------------------------------------------------------------
// LDS-tiled 3x3x3 pool (SAME padding). Boundary handling: replicate-clamp,
// which is exactly equivalent to -inf (max) / +inf (min) padding because
// every clamped OOB position maps to a border voxel already inside the
// window. Tile staged with GLOBAL_LOAD_ASYNC_TO_LDS_B32 (ASYNCcnt).
//   IS_MIN=true  : erode  -> writes dst
//   IS_MIN=false : dilate; DO_SKEL fuses skeleton update:
//     d = dilate(src); delta = relu(img - d);
//     first ? skel = delta : skel += max(1-skel,0)*delta
// Tile: 8(d) x 8(h) x 64(w) outputs, halo 1, 256 threads (8 waves).
// Per thread: one 16-wide W strip; 3x3 (d,h) column reduce + sliding-window
// ring over W -> ~10 LDS reads per output instead of 27.
// ---------------------------------------------------------------------------
template<bool IS_MIN, bool DO_SKEL>
__global__ __launch_bounds__(256) void pool3d(const float* __restrict__ src,
                                              const float* __restrict__ img,
                                              float* __restrict__ dst,
                                              int first) {
    __shared__ float tile[LDS_N];
    const int tw = blockIdx.x * TW;
    const int th = blockIdx.y * TH;
    const int td = (blockIdx.z & 15) * TD;
    const int bc = blockIdx.z >> 4;
    const size_t bcbase = (size_t)bc * VOL;
    const float* s = src + bcbase;
    const int tid = threadIdx.x;

    // --- stage (TD+2)x(TH+2)x(TW+2) tile via async copy, clamped coords ---
    for (int i = tid; i < LDS_N; i += 256) {
        int lw = i % LW;
        int t  = i / LW;
        int lh = t % LH;
        int ld = t / LH;
        int gw = min(max(tw + lw - 1, 0), W_ - 1);
        int gh = min(max(th + lh - 1, 0), H_ - 1);
        int gd = min(max(td + ld - 1, 0), D_ - 1);
        async_g2lds_b32(&s[(size_t)gd * SD_ + (size_t)gh * SH_ + gw], &tile[i]);
    }
    wait_asynccnt0();   // this wave's async copies landed in LDS
    __syncthreads();    // all waves' copies visible

    // --- compute: 256 threads = 4 W-strips x 8 H x 8 D ---
    const int wstrip = tid & 3;
    const int oh     = (tid >> 2) & 7;
    const int od     = tid >> 5;
    const int colbase = wstrip * 16;

    const size_t grow = bcbase + (size_t)(td + od) * SD_ +
                        (size_t)(th + oh) * SH_ + (size_t)(tw + colbase);

    float r0 = 0.0f, r1 = 0.0f;
#pragma unroll
    for (int j = 0; j < 18; ++j) {
        const int lwi = colbase + j;
        // 3x3 (d,h) column reduce at this w
        const float* c0 = &tile[(od + 0) * (LH * LW) + (oh + 0) * LW + lwi];
        const float* c1 = &tile[(od + 1) * (LH * LW) + (oh + 0) * LW + lwi];
        const float* c2 = &tile[(od + 2) * (LH * LW) + (oh + 0) * LW + lwi];
        float m = fminmax(c0[0], c0[LW], IS_MIN);
        m = fminmax(m, c0[2 * LW], IS_MIN);
        m = fminmax(m, fminmax(c1[0], c1[LW], IS_MIN), IS_MIN);
        m = fminmax(m, fminmax(c1[2 * LW], c2[0], IS_MIN), IS_MIN);
        m = fminmax(m, fminmax(c2[LW], c2[2 * LW], IS_MIN), IS_MIN);
        if (j >= 2) {
            float val = fminmax(fminmax(r0, r1, IS_MIN), m, IS_MIN);
            size_t g = grow + (size_t)(j - 2);
            if constexpr (DO_SKEL) {
                float im    = img[g];
                float delta = fmaxf(im - val, 0.0f);
                if (first) {
                    dst[g] = delta;
                } else {
                    float sk = dst[g];
                    dst[g] = sk + fmaxf(1.0f - sk, 0.0f) * delta;
                }
            } else {
                dst[g] = val;
            }
        }
        r0 = r1;
        r1 = m;
    }
}

// ---------------------------------------------------------------------------
// Per-(b,c) dual reduction: s0 = sum(sk*other), s1 = sum(sk).
// Wave fold via V_WMMA_F32_16X16X4_F32 with B = ones:
//   A(16x4): lane L VGPR0 = partial, VGPR1 = 0  =>  D[m][*] = p(m)+p(m+16)
//   sum 8 D VGPRs in-lane, then one __shfl_xor(16) completes the 32-lane sum.
// ---------------------------------------------------------------------------
__global__ __launch_bounds__(256) void k_reduce(const float* __restrict__ sk,
                                                const float* __restrict__ oth,
                                                float* __restrict__ parts) {
    const int bc = blockIdx.y;
    const float4* a = (const float4*)(sk  + (size_t)bc * VOL + (size_t)blockIdx.x * CHUNK);
    const float4* b = (const float4*)(oth + (size_t)bc * VOL + (size_t)blockIdx.x * CHUNK);
    float s0 = 0.0f, s1 = 0.0f;
    const int n4 = CHUNK / 4; /* 9216: 36 uniform iters per thread */
    for (int i = threadIdx.x; i < n4; i += 256) {
        __builtin_prefetch(&a[i + 2048], 0, 1); /* global_prefetch_b8 */
        __builtin_prefetch(&b[i + 2048], 0, 1);
        float4 x = a[i];
        float4 y = b[i];
        s0 = fmaf(x.x, y.x, s0);
        s0 = fmaf(x.y, y.y, s0);
        s0 = fmaf(x.z, y.z, s0);
        s0 = fmaf(x.w, y.w, s0);
        s1 += (x.x + x.y) + (x.z + x.w);
    }

    // Wave32 fold via f32 WMMA (EXEC is all-ones here: no divergence above).
    v2f a0 = {s0, 0.0f};
    v2f a1 = {s1, 0.0f};
    v2f on = {1.0f, 1.0f};
    v8f z  = {0.f, 0.f, 0.f, 0.f, 0.f, 0.f, 0.f, 0.f};
    v8f d0 = __builtin_amdgcn_wmma_f32_16x16x4_f32(false, a0, false, on,
                                                   (short)0, z, false, false);
    v8f d1 = __builtin_amdgcn_wmma_f32_16x16x4_f32(false, a1, false, on,
                                                   (short)0, z, false, false);
    float t0 = ((d0[0] + d0[1]) + (d0[2] + d0[3])) + ((d0[4] + d0[5]) + (d0[6] + d0[7]));
    float t1 = ((d1[0] + d1[1]) + (d1[2] + d1[3])) + ((d1[4] + d1[5]) + (d1[6] + d1[7]));
    t0 += __shfl_xor(t0, 16, 32);
    t1 += __shfl_xor(t1, 16, 32);

    __shared__ float ls[16];
    const int wv = threadIdx.x >> 5;
    if ((threadIdx.x & 31) == 0) {
        ls[wv]     = t0;
        ls[wv + 8] = t1;
    }
    __syncthreads();
    if (threadIdx.x == 0) {
        float u0 = 0.0f, u1 = 0.0f;
        for (int w = 0; w < 8; ++w) { u0 += ls[w]; u1 += ls[w + 8]; }
        parts[(size_t)bc * NBLK + blockIdx.x]          = u0; /* kind 0 */
        parts[(size_t)(BC_ + bc) * NBLK + blockIdx.x]  = u1; /* kind 1 */
    }
}

// ---------------------------------------------------------------------------
// Final scalar: parts layout [4 kinds][4 bc][NBLK]
//   kinds: 0=sum(sp*t) 1=sum(sp) 2=sum(st*p) 3=sum(st)
// ---------------------------------------------------------------------------
__global__ void k_finalize(const float* __restrict__ parts, float* __restrict__ out) {
    if (threadIdx.x != 0) return;
    float acc = 0.0f;
    for (int bc = 0; bc < BC_; ++bc) {
        float S[4];
        for (int k = 0; k < 4; ++k) {
            const float* p = parts + ((size_t)k * BC_ + bc) * NBLK;
            float s = 0.0f;
            for (int i = 0; i < NBLK; ++i) s += p[i];
            S[k] = s;
        }
        float tprec = (S[0] + 1.0f) / (S[1] + 1.0f);
        float tsens = (S[2] + 1.0f) / (S[3] + 1.0f);
        float cl    = 2.0f * tprec * tsens / (tprec + tsens + 1.0f);
        acc += 1.0f - cl;
    }
    out[0] = acc * 0.25f;
}

// ---------------------------------------------------------------------------
extern "C" void kernel_launch(void* const* d_in, const int* in_sizes, int n_in,
                              void* d_out, int out_size, void* d_ws, size_t ws_size,
                              hipStream_t stream) {
    const float* pred = (const float*)d_in[0];
    const float* targ = (const float*)d_in[1];
    float* wsf = (float*)d_ws;

    float* P     = wsf;            // sigmoid(pred)
    float* Ea    = P  + VOLS;      // erode ping
    float* Eb    = Ea + VOLS;      // erode pong
    float* SK    = Eb + VOLS;      // skeleton
    float* parts = SK + VOLS;      // [4][4][NBLK]

    dim3 pb(256);
    dim3 pg(W_ / TW, H_ / TH, (D_ / TD) * BC_); /* 3 x 24 x 64 */

    k_sigmoid<<<(unsigned)(VOLS / 4 / 256), 256, 0, stream>>>(
        (const float4*)pred, (float4*)P);

    for (int pass = 0; pass < 2; ++pass) {
        const float* X = (pass == 0) ? P : targ;
        const float* cur = X;
        // E_{k+1} = erode(E_k); open(E_k) = dilate(E_{k+1}); fused skel update
        for (int k = 0; k < 6; ++k) {
            float* nxt = (k & 1) ? Eb : Ea;
            pool3d<true,  false><<<pg, pb, 0, stream>>>(cur, nullptr, nxt, 0);
            pool3d<false, true ><<<pg, pb, 0, stream>>>(nxt, cur, SK, k == 0);
            cur = nxt;
        }
        const float* other = (pass == 0) ? targ : P;
        k_reduce<<<dim3(NBLK, BC_), 256, 0, stream>>>(
            SK, other, parts + (size_t)pass * 2 * BC_ * NBLK);
    }

    k_finalize<<<1, 1, 0, stream>>>(parts, (float*)d_out);
}